// Encoder_70557722739336
// MI455X (gfx1250) — compile-verified
//
#include <hip/hip_runtime.h>
#include <hip/hip_bf16.h>

typedef float v2f __attribute__((ext_vector_type(2)));
typedef float v8f __attribute__((ext_vector_type(8)));

constexpr int N_NODES = 100000;
constexpr int N_EDGES = 1000000;
constexpr int D       = 64;

// workspace layout (in floats)
constexpr size_t OFF_NORM_OUT = 0;        // 100000 floats (deg_out -> norm_out in place)
constexpr size_t OFF_NORM_IN  = 100352;   // 100000 floats (deg_in  -> norm_in  in place)
constexpr size_t OFF_AGG      = 200704;   // 100000*64 floats
constexpr size_t WS_FLOATS    = OFF_AGG + (size_t)N_NODES * D;   // 6,600,704 floats ~26.4MB

// Guaranteed hardware f32 atomic (global_atomic_add_f32, no return -> STOREcnt path)
__device__ __forceinline__ void atomAddF(float* p, float v) {
    unsafeAtomicAdd(p, v);
}

// ---------------------------------------------------------------- zero fill
__global__ void zero_kernel(float4* __restrict__ p, int n4) {
    int i = blockIdx.x * blockDim.x + threadIdx.x;
    if (i < n4) p[i] = make_float4(0.f, 0.f, 0.f, 0.f);
}

// ---------------------------------------------------------------- degrees
__global__ void degree_kernel(const int* __restrict__ src, const int* __restrict__ dst,
                              float* __restrict__ degOut, float* __restrict__ degIn,
                              int nEdges) {
    int e = blockIdx.x * blockDim.x + threadIdx.x;
    if (e >= nEdges) return;
    atomAddF(&degOut[src[e]], 1.0f);
    atomAddF(&degIn[dst[e]], 1.0f);
}

// deg -> rsqrt(max(deg,1)) in place (run over both arrays at once)
__global__ void norm_finalize_kernel(float* __restrict__ p, int n) {
    int i = blockIdx.x * blockDim.x + threadIdx.x;
    if (i < n) p[i] = rsqrtf(fmaxf(p[i], 1.0f));
}

// ---------------------------------------------------------------- edge scatter
// 16 threads per edge, 4 features each: gather x[src]*norm_out[src], atomic-add into agg[dst]
__global__ __launch_bounds__(256) void scatter_kernel(
        const int* __restrict__ src, const int* __restrict__ dst,
        const float* __restrict__ x, const float* __restrict__ normOut,
        float* __restrict__ agg, int nEdges) {
    int tid = blockIdx.x * blockDim.x + threadIdx.x;
    int e = tid >> 4;
    if (e >= nEdges) return;
    int f = (tid & 15) << 2;
    int s = src[e];
    int d = dst[e];
    float ns = normOut[s];
    float4 xv = *(const float4*)(x + (size_t)s * D + f);
    float* ap = agg + (size_t)d * D + f;
    atomAddF(ap + 0, xv.x * ns);
    atomAddF(ap + 1, xv.y * ns);
    atomAddF(ap + 2, xv.z * ns);
    atomAddF(ap + 3, xv.w * ns);
}

// ---------------------------------------------------------------- WMMA GEMM
// out[i,:] = act( (agg[i,:] * norm_in[i]) @ W + b ),  W:[64x64] row-major.
// One wave computes a 16-row x 64-col tile via V_WMMA_F32_16X16X4_F32.
// 256 threads/block = 8 waves = 8 row tiles per block.
// W staged into LDS in K-pair-interleaved layout: (k,n) -> sW[(k/2)*128 + 2n + (k&1)]
// so every B fragment (K, K+1 at fixed n) is one aligned ds_load_b64 into a v2f pair.
__global__ __launch_bounds__(256) void gemm_norm_bias_kernel(
        const float* __restrict__ agg, const float* __restrict__ normIn,
        const float* __restrict__ W, const float* __restrict__ bias,
        float* __restrict__ out, int nNodes, int applyRelu) {
    __shared__ float sW[D * D];
    __shared__ float sB[D];

    // cooperative stage of W with K-pair interleave; coalesced global reads
    for (int i = threadIdx.x; i < D * D; i += 256) {
        int k = i >> 6;
        int n = i & 63;
        sW[(k >> 1) * 128 + (n << 1) + (k & 1)] = W[i];
    }
    if (threadIdx.x < D) sB[threadIdx.x] = bias[threadIdx.x];
    __syncthreads();

    const int lane = threadIdx.x & 31;
    const int wave = threadIdx.x >> 5;
    const int tile = blockIdx.x * 8 + wave;    // 16-row tile index
    const int r0   = tile * 16;

    const int half = lane >> 4;   // 0: lanes 0-15, 1: lanes 16-31
    const int l16  = lane & 15;

    // A-fragment row for this lane (clamped so EXEC stays all-ones for WMMA)
    const int arow = min(r0 + l16, nNodes - 1);
    const float nrm = normIn[arow];
    const float* arowp = agg + (size_t)arow * D;

    v8f acc0 = {}, acc1 = {}, acc2 = {}, acc3 = {};

    #pragma unroll
    for (int k0 = 0; k0 < D; k0 += 4) {
        // A 16x4 f32 frag: VGPR0 holds K = k0 + 2*half, VGPR1 holds K+1
        const int ka = k0 + 2 * half;
        v2f a = *(const v2f*)(arowp + ka);
        a.x *= nrm;
        a.y *= nrm;

        // B 4x16 f32 frags: pair row (k0+2*half)/2 = k0/2 + half
        const float* bp = sW + (size_t)((k0 >> 1) + half) * 128;
        v2f b0 = *(const v2f*)(bp + (( 0 + l16) << 1));
        v2f b1 = *(const v2f*)(bp + ((16 + l16) << 1));
        v2f b2 = *(const v2f*)(bp + ((32 + l16) << 1));
        v2f b3 = *(const v2f*)(bp + ((48 + l16) << 1));

#if defined(__gfx1250__) && __has_builtin(__builtin_amdgcn_wmma_f32_16x16x4_f32)
        acc0 = __builtin_amdgcn_wmma_f32_16x16x4_f32(false, a, false, b0, (short)0, acc0, false, false);
        acc1 = __builtin_amdgcn_wmma_f32_16x16x4_f32(false, a, false, b1, (short)0, acc1, false, false);
        acc2 = __builtin_amdgcn_wmma_f32_16x16x4_f32(false, a, false, b2, (short)0, acc2, false, false);
        acc3 = __builtin_amdgcn_wmma_f32_16x16x4_f32(false, a, false, b3, (short)0, acc3, false, false);
#else
        // fallback to keep non-gfx1250 compiles alive (never the intended path)
        acc0[0] += a.x * b0.x; acc1[0] += a.x * b1.x;
        acc2[0] += a.x * b2.x; acc3[0] += a.x * b3.x;
#endif
    }

    // epilogue: bias + optional relu + store.  C/D layout: VGPR v, lane -> row r0+v+8*half, col l16
    const float bc0 = sB[ 0 + l16];
    const float bc1 = sB[16 + l16];
    const float bc2 = sB[32 + l16];
    const float bc3 = sB[48 + l16];

    #pragma unroll
    for (int v = 0; v < 8; ++v) {
        int row = r0 + v + 8 * half;
        if (row < nNodes) {
            float o0 = acc0[v] + bc0;
            float o1 = acc1[v] + bc1;
            float o2 = acc2[v] + bc2;
            float o3 = acc3[v] + bc3;
            if (applyRelu) {
                o0 = fmaxf(o0, 0.f); o1 = fmaxf(o1, 0.f);
                o2 = fmaxf(o2, 0.f); o3 = fmaxf(o3, 0.f);
            }
            float* op = out + (size_t)row * D;
            op[ 0 + l16] = o0;
            op[16 + l16] = o1;
            op[32 + l16] = o2;
            op[48 + l16] = o3;
        }
    }
}

// ---------------------------------------------------------------- launcher
extern "C" void kernel_launch(void* const* d_in, const int* in_sizes, int n_in,
                              void* d_out, int out_size, void* d_ws, size_t ws_size,
                              hipStream_t stream) {
    const int*   src     = (const int*)d_in[0];
    const int*   dst     = (const int*)d_in[1];
    const float* in_feat = (const float*)d_in[2];
    const float* W1      = (const float*)d_in[3];
    const float* b1      = (const float*)d_in[4];
    const float* W2      = (const float*)d_in[5];
    const float* b2      = (const float*)d_in[6];
    float* out = (float*)d_out;

    float* ws       = (float*)d_ws;
    float* normOut  = ws + OFF_NORM_OUT;
    float* normIn   = ws + OFF_NORM_IN;
    float* agg      = ws + OFF_AGG;

    const int tilesBlocks = ((N_NODES / 16) + 7) / 8;   // 6250 tiles / 8 waves = 782 blocks

    // 1) zero norms + agg (whole used ws region)
    {
        int n4 = (int)(WS_FLOATS / 4);
        zero_kernel<<<(n4 + 255) / 256, 256, 0, stream>>>((float4*)ws, n4);
    }
    // 2) degrees (into norm buffers), 3) finalize norms
    degree_kernel<<<(N_EDGES + 255) / 256, 256, 0, stream>>>(src, dst, normOut, normIn, N_EDGES);
    norm_finalize_kernel<<<((int)OFF_AGG + 255) / 256, 256, 0, stream>>>(ws, (int)OFF_AGG);

    // ---- layer 1: in_feat -> h (stored in d_out) ----
    {
        int nThreads = N_EDGES * 16;
        scatter_kernel<<<(nThreads + 255) / 256, 256, 0, stream>>>(src, dst, in_feat, normOut, agg, N_EDGES);
        gemm_norm_bias_kernel<<<tilesBlocks, 256, 0, stream>>>(agg, normIn, W1, b1, out, N_NODES, 1);
    }

    // ---- layer 2: h (d_out) -> d_out ----
    {
        int n4 = (N_NODES * D) / 4;
        zero_kernel<<<(n4 + 255) / 256, 256, 0, stream>>>((float4*)agg, n4);
        int nThreads = N_EDGES * 16;
        scatter_kernel<<<(nThreads + 255) / 256, 256, 0, stream>>>(src, dst, out, normOut, agg, N_EDGES);
        gemm_norm_bias_kernel<<<tilesBlocks, 256, 0, stream>>>(agg, normIn, W2, b2, out, N_NODES, 0);
    }
}